// Encoder_GCN_70428873720344
// MI455X (gfx1250) — compile-verified
//
#include <hip/hip_runtime.h>

// GCN encoder: h = x@Wgc ; agg = scatter_add(h[src]*norm, dst) + bgc ; out = agg@Wfc + bfc
// N=100000, E=1600000, IN=256, HID=128, OUT=64. All fp32 (exact: fp32 WMMA 16x16x4).

#define N_IN  256
#define N_HID 128
#define N_OUT 64

typedef __attribute__((ext_vector_type(2))) float v2f;
typedef __attribute__((ext_vector_type(8))) float v8f;

// ---- degree / normalization -------------------------------------------------

__global__ void k_init_deg(float* __restrict__ deg, int n) {
  int i = blockIdx.x * blockDim.x + threadIdx.x;
  if (i < n) deg[i] = 1.0f;                      // self-loop contributes 1
}

__global__ void k_deg_accum(const int* __restrict__ ei, float* __restrict__ deg, int E) {
  int e = blockIdx.x * blockDim.x + threadIdx.x;
  if (e < E) atomicAdd(&deg[ei[E + e]], 1.0f);   // dst = ei[1][e]
}

__global__ void k_dinv(float* __restrict__ deg, int n) {
  int i = blockIdx.x * blockDim.x + threadIdx.x;
  if (i < n) {
    float d = deg[i];
    deg[i] = (d > 0.0f) ? rsqrtf(d) : 0.0f;      // in-place: deg -> dinv
  }
}

// ---- GEMM1: h[N,128] = x[N,256] @ Wgc[256,128]  (fp32 WMMA 16x16x4) ---------
// One wave handles a 16-row tile x all 8 column tiles of 16.

__global__ __launch_bounds__(256) void k_gemm1(const float* __restrict__ x,
                                               const float* __restrict__ W,
                                               float* __restrict__ h, int n) {
  int wave = (blockIdx.x * blockDim.x + threadIdx.x) >> 5;
  int row0 = wave * 16;
  if (row0 >= n) return;                          // uniform per wave
  int lane = threadIdx.x & 31;
  int half = lane >> 4;                           // 0: K even pair base 0, 1: base 2
  int lrow = lane & 15;

  v8f acc[8] = {};
  int arow = row0 + lrow; if (arow >= n) arow = n - 1;
  const float* ar = x + (size_t)arow * N_IN + 2 * half;

  for (int k = 0; k < N_IN; k += 4) {
    v2f a; a.x = ar[k]; a.y = ar[k + 1];          // A: K = k+2*half, k+2*half+1
    const float* b0 = W + (size_t)(k + 2 * half) * N_HID + lrow;
#pragma unroll
    for (int c = 0; c < 8; ++c) {
      v2f b; b.x = b0[c * 16]; b.y = b0[N_HID + c * 16];
      acc[c] = __builtin_amdgcn_wmma_f32_16x16x4_f32(
          false, a, false, b, (short)0, acc[c], false, false);
    }
  }
  // D layout: VGPR r -> rows {row0+r, row0+r+8}; col = c*16 + lrow
  float* hr = h + (size_t)(row0 + 8 * half) * N_HID + lrow;
#pragma unroll
  for (int c = 0; c < 8; ++c)
#pragma unroll
    for (int r = 0; r < 8; ++r)
      if (row0 + 8 * half + r < n)
        hr[(size_t)r * N_HID + c * 16] = acc[c][r];
}

// ---- agg init: agg[i,f] = bgc[f] + h[i,f]*dinv[i]^2  (self-loop term) -------

__global__ void k_init_agg(const float* __restrict__ h, const float* __restrict__ dinv,
                           const float* __restrict__ bgc, float* __restrict__ agg, int n) {
  long long idx = (long long)blockIdx.x * blockDim.x + threadIdx.x;
  if (idx >= (long long)n * N_HID) return;
  int i = (int)(idx >> 7);
  int f = (int)(idx & (N_HID - 1));
  float w = dinv[i];
  agg[idx] = bgc[f] + h[idx] * w * w;
}

// ---- edge scatter: agg[dst] += h[src] * dinv[src]*dinv[dst] -----------------
// 128 threads per edge: coalesced 512B row gather, f32 atomic scatter (L2-resident agg).

__global__ void k_scatter(const float* __restrict__ h, const float* __restrict__ dinv,
                          const int* __restrict__ ei, float* __restrict__ agg, int E) {
  long long idx = (long long)blockIdx.x * blockDim.x + threadIdx.x;
  if (idx >= (long long)E * N_HID) return;
  int e = (int)(idx >> 7);
  int f = (int)(idx & (N_HID - 1));
  int s = ei[e];
  int d = ei[E + e];
  float w = dinv[s] * dinv[d];
  atomicAdd(&agg[(size_t)d * N_HID + f], h[(size_t)s * N_HID + f] * w);
}

// ---- GEMM2: out[N,64] = agg[N,128] @ Wfc[128,64] + bfc ----------------------

__global__ __launch_bounds__(256) void k_gemm2(const float* __restrict__ agg,
                                               const float* __restrict__ W,
                                               const float* __restrict__ bias,
                                               float* __restrict__ out, int n) {
  int wave = (blockIdx.x * blockDim.x + threadIdx.x) >> 5;
  int row0 = wave * 16;
  if (row0 >= n) return;
  int lane = threadIdx.x & 31;
  int half = lane >> 4;
  int lrow = lane & 15;

  v8f acc[4] = {};
  int arow = row0 + lrow; if (arow >= n) arow = n - 1;
  const float* ar = agg + (size_t)arow * N_HID + 2 * half;

  for (int k = 0; k < N_HID; k += 4) {
    v2f a; a.x = ar[k]; a.y = ar[k + 1];
    const float* b0 = W + (size_t)(k + 2 * half) * N_OUT + lrow;
#pragma unroll
    for (int c = 0; c < 4; ++c) {
      v2f b; b.x = b0[c * 16]; b.y = b0[N_OUT + c * 16];
      acc[c] = __builtin_amdgcn_wmma_f32_16x16x4_f32(
          false, a, false, b, (short)0, acc[c], false, false);
    }
  }
  float* orow = out + (size_t)(row0 + 8 * half) * N_OUT + lrow;
#pragma unroll
  for (int c = 0; c < 4; ++c) {
    float bv = bias[c * 16 + lrow];
#pragma unroll
    for (int r = 0; r < 8; ++r)
      if (row0 + 8 * half + r < n)
        orow[(size_t)r * N_OUT + c * 16] = acc[c][r] + bv;
  }
}

// ---- launcher ---------------------------------------------------------------

extern "C" void kernel_launch(void* const* d_in, const int* in_sizes, int n_in,
                              void* d_out, int out_size, void* d_ws, size_t ws_size,
                              hipStream_t stream) {
  const float* x   = (const float*)d_in[0];
  const int*   ei  = (const int*)d_in[1];    // [2, E] row-major: src = ei[e], dst = ei[E+e]
  const float* Wgc = (const float*)d_in[2];
  const float* bgc = (const float*)d_in[3];
  const float* Wfc = (const float*)d_in[4];
  const float* bfc = (const float*)d_in[5];
  float* out = (float*)d_out;

  const int n = in_sizes[0] / N_IN;          // 100000
  const int E = in_sizes[1] / 2;             // 1600000

  // workspace: dinv [n] | h [n*128] | agg [n*128]  (~103 MB)
  float* base = (float*)d_ws;
  float* dinv = base;
  float* h    = base + 131072;               // 512KB-aligned region for dinv
  float* agg  = h + (size_t)n * N_HID;

  const int T = 256;
  k_init_deg <<<(n + T - 1) / T, T, 0, stream>>>(dinv, n);
  k_deg_accum<<<(E + T - 1) / T, T, 0, stream>>>(ei, dinv, E);
  k_dinv     <<<(n + T - 1) / T, T, 0, stream>>>(dinv, n);

  int waves  = (n + 15) / 16;                // 6250 wave-tiles
  int blocks = (waves + 7) / 8;              // 8 waves (256 thr) per block
  k_gemm1<<<blocks, T, 0, stream>>>(x, Wgc, h, n);

  long long tot1 = (long long)n * N_HID;
  k_init_agg<<<(int)((tot1 + T - 1) / T), T, 0, stream>>>(h, dinv, bgc, agg, n);

  long long tot2 = (long long)E * N_HID;
  k_scatter<<<(int)((tot2 + T - 1) / T), T, 0, stream>>>(h, dinv, ei, agg, E);

  k_gemm2<<<blocks, T, 0, stream>>>(agg, Wfc, bfc, out, n);
}